// GCNGRUBaseline_75118978007588
// MI455X (gfx1250) — compile-verified
//
#include <hip/hip_runtime.h>
#include <hip/hip_bf16.h>
#include <stdint.h>

typedef __attribute__((ext_vector_type(2))) float v2f;
typedef __attribute__((ext_vector_type(8))) float v8f;

#define FH 64    // hidden features
#define FIN 32   // input features
#define TSTEPS 12
#define KBMAX 16  // max Fin/4 (Fin <= 64)
#define SLAB 80   // (kb,hi)-slab stride in float2: 160 words == 32 mod 64 banks

// ---------------------------------------------------------------- utilities
__global__ void fill_f32(float* __restrict__ p, float v, long n) {
  long i = (long)blockIdx.x * blockDim.x + threadIdx.x;
  if (i < n) p[i] = v;
}

__global__ void neg_copy(float* __restrict__ out, const float* __restrict__ in, long n) {
  long i = (long)blockIdx.x * blockDim.x + threadIdx.x;
  if (i < n) out[i] = -in[i];
}

__global__ void copy_f32(float* __restrict__ out, const float* __restrict__ in, long n) {
  long i = (long)blockIdx.x * blockDim.x + threadIdx.x;
  if (i < n) out[i] = in[i];
}

// ----------------------------------------------------- per-timestep edge prep
__global__ void edge_prep(const long long* __restrict__ s64, const long long* __restrict__ d64,
                          const float* __restrict__ ew, int* __restrict__ src,
                          int* __restrict__ dst, float* __restrict__ deg, int E) {
  int e = blockIdx.x * blockDim.x + threadIdx.x;
  if (e < E) {
    int s = (int)s64[e];
    int d = (int)d64[e];
    src[e] = s;
    dst[e] = d;
    atomicAdd(deg + s, ew[e]);
  }
}

__global__ void deg_to_dis(float* __restrict__ deg, int N) {
  int i = blockIdx.x * blockDim.x + threadIdx.x;
  if (i < N) {
    float d = deg[i];
    deg[i] = (d > 0.f) ? rsqrtf(d) : 0.f;
  }
}

__global__ void edge_norm(const float* __restrict__ ew, const int* __restrict__ src,
                          const int* __restrict__ dst, const float* __restrict__ dis,
                          float* __restrict__ norm, int E) {
  int e = blockIdx.x * blockDim.x + threadIdx.x;
  if (e < E) norm[e] = -ew[e] * dis[src[e]] * dis[dst[e]];
}

// out[dst, 4q..4q+3] += scale * norm[e] * v[src, 4q..4q+3]; logQ = log2(F/4)
__global__ void prop_scatter4(const float4* __restrict__ v, const int* __restrict__ src,
                              const int* __restrict__ dst, const float* __restrict__ norm,
                              float* __restrict__ out, float scale, long total, int logQ) {
  long gid = (long)blockIdx.x * blockDim.x + threadIdx.x;
  if (gid < total) {
    int qmask = (1 << logQ) - 1;
    long e = gid >> logQ;
    int q = (int)(gid & qmask);
    float w = scale * norm[e];
    float4 val = v[(((long)src[e]) << logQ) + q];
    float* o = out + (((((long)dst[e]) << logQ) + q) << 2);
    atomicAdd(o + 0, w * val.x);
    atomicAdd(o + 1, w * val.y);
    atomicAdd(o + 2, w * val.z);
    atomicAdd(o + 3, w * val.w);
  }
}

// ------------------------------------------------------------- WMMA cheb GEMM
// Fused multi-output Chebyshev GEMM:
//   Out_o[n,:] = (acc ? Out_o[n,:] : 0) + bias_o + sum_k A_k @ W_o[k]
// One wave owns a 16x64 tile per output (NOUT x 4 v8f accumulators).
// Current-k weights staged in LDS in fragment-ready float2 form:
//   sW[((o*KBMAX+kb)*2+hi)*SLAB + col] = { W[kb*4+hi*2][col], W[kb*4+hi*2+1][col] }
// so each B fragment is a single aligned ds_load_b64 straight into a WMMA pair.
template <int NOUT>
__global__ void cheb_mm_multi(const float* __restrict__ A0, const float* __restrict__ A1,
                              const float* __restrict__ A2,
                              const float* __restrict__ W0, const float* __restrict__ b0, float* __restrict__ O0,
                              const float* __restrict__ W1, const float* __restrict__ b1, float* __restrict__ O1,
                              const float* __restrict__ W2, const float* __restrict__ b2, float* __restrict__ O2,
                              int N, int Fin, int accFlag) {
  __shared__ float2 sW[NOUT * KBMAX * 2 * SLAB];

  const float* W[3] = {W0, W1, W2};
  const float* bias[3] = {b0, b1, b2};
  float* Out[3] = {O0, O1, O2};

  int lane = threadIdx.x & 31;
  int wave = threadIdx.x >> 5;
  int tile = blockIdx.x * (blockDim.x >> 5) + wave;
  int ntiles = (N + 15) >> 4;
  if (tile >= ntiles) tile = ntiles - 1;   // keep waves alive for barriers; dup stores benign
  int m0 = tile << 4;
  int l15 = lane & 15;
  int hi = lane >> 4;

  // accumulator init: bias (+ existing output if accumulating)
  v8f acc[NOUT][4];
#pragma unroll
  for (int o = 0; o < NOUT; ++o) {
#pragma unroll
    for (int c = 0; c < 4; ++c) {
      int col = c * 16 + l15;
      float bv = bias[o][col];
#pragma unroll
      for (int j = 0; j < 8; ++j) {
        int r = m0 + j + hi * 8;
        float cv = bv;
        if (accFlag) {
          int rr = (r < N) ? r : (N - 1);
          cv += Out[o][(long)rr * FH + col];
        }
        acc[o][c][j] = cv;
      }
    }
  }

  int rowA = m0 + l15;
  if (rowA >= N) rowA = N - 1;            // clamp; stores are guarded

  const int kbN = Fin >> 2;               // k-blocks of 4
  const int stageN = kbN * 2 * 64;        // float2 entries per (o,k) slab

#pragma unroll
  for (int k = 0; k < 3; ++k) {
    if (k) __syncthreads();               // previous compute done before overwrite
    // cooperative fragment-ready staging of W_o[k]
#pragma unroll
    for (int o = 0; o < NOUT; ++o) {
      const float* Wg = W[o] + (long)k * Fin * FH;
      for (int i = (int)threadIdx.x; i < stageN; i += (int)blockDim.x) {
        int col = i & 63;
        int rest = i >> 6;
        int hh = rest & 1;
        int kb = rest >> 1;
        int krow = kb * 4 + hh * 2;
        float2 vv;
        vv.x = Wg[krow * FH + col];
        vv.y = Wg[(krow + 1) * FH + col];
        sW[((o * KBMAX + kb) * 2 + hh) * SLAB + col] = vv;
      }
    }
    __syncthreads();

    const float* arow = (k == 0 ? A0 : (k == 1 ? A1 : A2)) + (long)rowA * Fin;
    for (int kb = 0; kb < kbN; ++kb) {
      // A 16x4 fragment: VGPR0 = K {0,2}, VGPR1 = K {1,3} across lane halves
      int kbase = kb * 4 + hi * 2;
      v2f a;
      a.x = arow[kbase];
      a.y = arow[kbase + 1];
#pragma unroll
      for (int o = 0; o < NOUT; ++o) {
        const float2* frag = &sW[((o * KBMAX + kb) * 2 + hi) * SLAB];
#pragma unroll
        for (int c = 0; c < 4; ++c) {
          float2 bb = frag[c * 16 + l15];  // single ds_load_b64 -> WMMA pair
          v2f b;
          b.x = bb.x;
          b.y = bb.y;
          acc[o][c] = __builtin_amdgcn_wmma_f32_16x16x4_f32(
              false, a, false, b, (short)0, acc[o][c], false, false);
        }
      }
    }
  }

#pragma unroll
  for (int o = 0; o < NOUT; ++o) {
#pragma unroll
    for (int c = 0; c < 4; ++c) {
      int col = c * 16 + l15;
#pragma unroll
      for (int j = 0; j < 8; ++j) {
        int r = m0 + j + hi * 8;
        if (r < N) Out[o][(long)r * FH + col] = acc[o][c][j];
      }
    }
  }
}

// ------------------------------------------------------------- GRU elementwise
__global__ void gru_zr(float* __restrict__ Xz, const float* __restrict__ Xr,
                       const float* __restrict__ h, float* __restrict__ hr, long n) {
  long i = (long)blockIdx.x * blockDim.x + threadIdx.x;
  if (i < n) {
    float z = 1.f / (1.f + expf(-Xz[i]));
    float r = 1.f / (1.f + expf(-Xr[i]));
    Xz[i] = z;
    hr[i] = h[i] * r;
  }
}

__global__ void gru_update(float* __restrict__ h, const float* __restrict__ z,
                           const float* __restrict__ Xh, long n) {
  long i = (long)blockIdx.x * blockDim.x + threadIdx.x;
  if (i < n) {
    float zt = z[i];
    float ht = tanhf(Xh[i]);
    h[i] = zt * h[i] + (1.f - zt) * ht;
  }
}

__global__ void head_kernel(const float* __restrict__ h, const float* __restrict__ Wlin,
                            const float* __restrict__ blin, float* __restrict__ out, int N) {
  int n = blockIdx.x * blockDim.x + threadIdx.x;
  if (n < N) {
    float s = blin[0];
    const float4* hv = (const float4*)(h + (long)n * FH);
    const float4* wv = (const float4*)Wlin;
#pragma unroll
    for (int f = 0; f < FH / 4; ++f) {
      float4 a = hv[f], b = wv[f];
      s += a.x * b.x + a.y * b.y + a.z * b.z + a.w * b.w;
    }
    out[n] = s;
  }
}

// ------------------------------------------------------------------- launcher
static inline int nblk(long n, int b) { return (int)((n + b - 1) / b); }

extern "C" void kernel_launch(void* const* d_in, const int* in_sizes, int n_in,
                              void* d_out, int out_size, void* d_ws, size_t ws_size,
                              hipStream_t stream) {
  const float* x      = (const float*)d_in[0];
  const long long* ei = (const long long*)d_in[1];
  const float* ew     = (const float*)d_in[2];
  const float* Wxz = (const float*)d_in[3];  const float* bxz = (const float*)d_in[4];
  const float* Whz = (const float*)d_in[5];  const float* bhz = (const float*)d_in[6];
  const float* Wxr = (const float*)d_in[7];  const float* bxr = (const float*)d_in[8];
  const float* Whr = (const float*)d_in[9];  const float* bhr = (const float*)d_in[10];
  const float* Wxh = (const float*)d_in[11]; const float* bxh = (const float*)d_in[12];
  const float* Whh = (const float*)d_in[13]; const float* bhh = (const float*)d_in[14];
  const float* Wlin = (const float*)d_in[15];
  const float* blin = (const float*)d_in[16];

  const long N = (long)in_sizes[0] / ((long)TSTEPS * FIN);
  const long E = (long)in_sizes[2] / TSTEPS;

  // workspace carve (256B aligned)
  char* w = (char*)d_ws;
  auto carve = [&](size_t bytes) {
    void* p = (void*)w;
    w += (bytes + 255) & ~(size_t)255;
    return p;
  };
  float* deg  = (float*)carve(N * 4);            // becomes dis in place
  float* nrm  = (float*)carve(E * 4);
  int*   src  = (int*)carve(E * 4);
  int*   dst  = (int*)carve(E * 4);
  float* h    = (float*)carve(N * FH * 4);
  float* Tx1  = (float*)carve(N * FIN * 4);
  float* Tx2  = (float*)carve(N * FIN * 4);
  float* Th1  = (float*)carve(N * FH * 4);
  float* Th2  = (float*)carve(N * FH * 4);
  float* hr   = (float*)carve(N * FH * 4);
  float* Thr1 = (float*)carve(N * FH * 4);
  float* Thr2 = (float*)carve(N * FH * 4);
  float* Xz   = (float*)carve(N * FH * 4);
  float* Xr   = (float*)carve(N * FH * 4);
  float* Xh   = (float*)carve(N * FH * 4);

  const int B = 256;
  const long nh = N * FH;
  const long nx = N * FIN;
  const int ntiles = (int)((N + 15) >> 4);
  const int mmBlocks = (ntiles + 7) / 8;        // 8 waves (16-row tiles) per block
  const long sx4 = E * (FIN / 4);               // scatter quads, x (F=32 -> logQ=3)
  const long sh4 = E * (FH / 4);                // scatter quads, h (F=64 -> logQ=4)

  fill_f32<<<nblk(nh, B), B, 0, stream>>>(h, 0.f, nh);

  for (int t = 0; t < TSTEPS; ++t) {
    const float* xt = x + (size_t)t * nx;
    const long long* s64 = ei + (size_t)t * 2 * E;
    const long long* d64 = s64 + E;
    const float* ewt = ew + (size_t)t * E;

    // normalization
    fill_f32<<<nblk(N, B), B, 0, stream>>>(deg, 0.f, N);
    edge_prep<<<nblk(E, B), B, 0, stream>>>(s64, d64, ewt, src, dst, deg, (int)E);
    deg_to_dis<<<nblk(N, B), B, 0, stream>>>(deg, (int)N);
    edge_norm<<<nblk(E, B), B, 0, stream>>>(ewt, src, dst, deg, nrm, (int)E);

    // Chebyshev basis for x: Tx1 = L~ x, Tx2 = 2 L~ Tx1 - x
    fill_f32<<<nblk(nx, B), B, 0, stream>>>(Tx1, 0.f, nx);
    prop_scatter4<<<nblk(sx4, B), B, 0, stream>>>((const float4*)xt, src, dst, nrm, Tx1, 1.f, sx4, 3);
    neg_copy<<<nblk(nx, B), B, 0, stream>>>(Tx2, xt, nx);
    prop_scatter4<<<nblk(sx4, B), B, 0, stream>>>((const float4*)Tx1, src, dst, nrm, Tx2, 2.f, sx4, 3);

    // Chebyshev basis for h
    fill_f32<<<nblk(nh, B), B, 0, stream>>>(Th1, 0.f, nh);
    prop_scatter4<<<nblk(sh4, B), B, 0, stream>>>((const float4*)h, src, dst, nrm, Th1, 1.f, sh4, 4);
    neg_copy<<<nblk(nh, B), B, 0, stream>>>(Th2, h, nh);
    prop_scatter4<<<nblk(sh4, B), B, 0, stream>>>((const float4*)Th1, src, dst, nrm, Th2, 2.f, sh4, 4);

    // gate pre-activations: x path (3 outputs), h path (2 outputs, accumulate)
    cheb_mm_multi<3><<<mmBlocks, B, 0, stream>>>(xt, Tx1, Tx2,
                                                 Wxz, bxz, Xz,
                                                 Wxr, bxr, Xr,
                                                 Wxh, bxh, Xh, (int)N, FIN, 0);
    cheb_mm_multi<2><<<mmBlocks, B, 0, stream>>>(h, Th1, Th2,
                                                 Whz, bhz, Xz,
                                                 Whr, bhr, Xr,
                                                 nullptr, nullptr, nullptr, (int)N, FH, 1);

    // z, r, h*r
    gru_zr<<<nblk(nh, B), B, 0, stream>>>(Xz, Xr, h, hr, nh);

    // Chebyshev basis for h*r and candidate term (accumulate into Xh)
    fill_f32<<<nblk(nh, B), B, 0, stream>>>(Thr1, 0.f, nh);
    prop_scatter4<<<nblk(sh4, B), B, 0, stream>>>((const float4*)hr, src, dst, nrm, Thr1, 1.f, sh4, 4);
    neg_copy<<<nblk(nh, B), B, 0, stream>>>(Thr2, hr, nh);
    prop_scatter4<<<nblk(sh4, B), B, 0, stream>>>((const float4*)Thr1, src, dst, nrm, Thr2, 2.f, sh4, 4);
    cheb_mm_multi<1><<<mmBlocks, B, 0, stream>>>(hr, Thr1, Thr2,
                                                 Whh, bhh, Xh,
                                                 nullptr, nullptr, nullptr,
                                                 nullptr, nullptr, nullptr, (int)N, FH, 1);

    // h = z*h + (1-z)*tanh(Xh)
    gru_update<<<nblk(nh, B), B, 0, stream>>>(h, Xz, Xh, nh);
  }

  // outputs: [ out(:,0) | h ]
  float* out = (float*)d_out;
  head_kernel<<<nblk(N, B), B, 0, stream>>>(h, Wlin, blin, out, (int)N);
  copy_f32<<<nblk(nh, B), B, 0, stream>>>(out + N, h, nh);
}